// GAT_83889301226057
// MI455X (gfx1250) — compile-verified
//
#include <hip/hip_runtime.h>
#include <math.h>

typedef float v2f __attribute__((ext_vector_type(2)));
typedef float v8f __attribute__((ext_vector_type(8)));
typedef unsigned int u32x4 __attribute__((ext_vector_type(4)));
typedef int i32x4 __attribute__((ext_vector_type(4)));
typedef int i32x8 __attribute__((ext_vector_type(8)));

#define N_NODES 50000
#define N_EDGES 800000
#define F_IN    128
#define EDIM    16
#define HID     64
#define HC1     128
#define NGRAPH  50
#define NCLS    2

// ---------------------------------------------------------------------------
// CAS-based float atomic max (handles -inf init correctly)
// ---------------------------------------------------------------------------
__device__ __forceinline__ void atomicMaxF(float* addr, float val) {
    int* ai = (int*)addr;
    int old = __float_as_int(*addr);
    while (__int_as_float(old) < val) {
        int assumed = old;
        old = atomicCAS(ai, assumed, __float_as_int(val));
        if (old == assumed) break;
    }
}

// ---------------------------------------------------------------------------
// Stage NFLOATS contiguous floats from global into LDS, block-cooperative.
// Preferred path: CDNA5 Tensor Data Mover (tensor_load_to_lds, TENSORcnt).
// Toolchain here exposes the 6-arg form: (g0, g1, g2, g3, g_extra, cpol).
// Fallback 1: async global->LDS copies (ASYNCcnt).
// Fallback 2: plain loads. All paths end with a full block barrier.
// ---------------------------------------------------------------------------
template <int NFLOATS>
__device__ __forceinline__ void stage_to_lds(float* __restrict__ sdst,
                                             const float* __restrict__ gsrc) {
#if defined(__gfx1250__) && __has_builtin(__builtin_amdgcn_tensor_load_to_lds)
    if (threadIdx.x < 32) {   // one wave issues the TDM descriptor
        unsigned long long ga = (unsigned long long)(const void*)gsrc;
        unsigned lds_off = (unsigned)(unsigned long long)sdst;  // addr[31:0] == LDS offset
        // D# group 0: count=1, lds_addr, 57-bit global addr, type=2 ("image")
        u32x4 g0;
        g0.x = 1u;
        g0.y = lds_off;
        g0.z = (unsigned)(ga & 0xffffffffu);
        g0.w = (unsigned)((ga >> 32) & 0x1ffffffu) | (2u << 30);
        // D# group 1: wgmask=0, data_size=4B (code 2), 1-D tile of NFLOATS elems
        i32x8 g1;
        g1[0] = (2 << 16);                                    // data_size = 4 bytes
        g1[1] = (int)((NFLOATS & 0xffff) << 16);              // tensor_dim0[15:0]
        g1[2] = (int)(((NFLOATS >> 16) & 0xffff) | (1 << 16));// tensor_dim0[31:16], tensor_dim1=1
        g1[3] = (int)((NFLOATS & 0xffff) << 16);              // tile_dim0 = NFLOATS
        g1[4] = 1;                                            // tile_dim1 = 1, tile_dim2 = 0
        g1[5] = NFLOATS;                                      // tensor_dim0_stride[31:0]
        g1[6] = 0;
        g1[7] = 0;
        i32x4 z4 = {0, 0, 0, 0};
        i32x8 z8 = {0, 0, 0, 0, 0, 0, 0, 0};
        __builtin_amdgcn_tensor_load_to_lds(g0, g1, z4, z4, z8, 0);
#if __has_builtin(__builtin_amdgcn_s_wait_tensorcnt)
        __builtin_amdgcn_s_wait_tensorcnt(0);
#else
        asm volatile("s_wait_tensorcnt 0x0" ::: "memory");
#endif
    }
    __syncthreads();
#elif defined(__gfx1250__) && __has_builtin(__builtin_amdgcn_global_load_async_to_lds_b128)
    for (int i = threadIdx.x * 4; i < NFLOATS; i += blockDim.x * 4)
        __builtin_amdgcn_global_load_async_to_lds_b128(
            (const __attribute__((address_space(1))) void*)(unsigned long long)(gsrc + i),
            (__attribute__((address_space(3))) void*)(unsigned)(unsigned long long)(sdst + i),
            0, 0);
#if __has_builtin(__builtin_amdgcn_s_wait_asynccnt)
    __builtin_amdgcn_s_wait_asynccnt(0);
#else
    asm volatile("s_wait_asynccnt 0x0" ::: "memory");
#endif
    __syncthreads();
#else
    for (int i = threadIdx.x; i < NFLOATS; i += blockDim.x) sdst[i] = gsrc[i];
    __syncthreads();
#endif
}

// ---------------------------------------------------------------------------
// fill kernel (workspace init; ws is poisoned by harness)
// ---------------------------------------------------------------------------
__global__ void fill_kernel(float* __restrict__ p, float v, long long n) {
    long long i = (long long)blockIdx.x * blockDim.x + threadIdx.x;
    long long s = (long long)gridDim.x * blockDim.x;
    for (; i < n; i += s) p[i] = v;
}

// ---------------------------------------------------------------------------
// WMMA fp32 GEMM:  C[M x Nc] = A[M x K] @ W[K x Nc] + bias, W staged in LDS.
// Block = 256 threads = 8 waves; block covers a 128-row M panel x full Nc.
// Each wave holds Nc/16 accumulator tiles; A-fragment loaded once per k-step
// and reused across all N-tiles (Nc/16 back-to-back v_wmma per step).
// M % 16 == 0 (50000 = 3125*16); K % 4 == 0; Nc % 16 == 0.
// ---------------------------------------------------------------------------
template <int K, int Nc>
__global__ __launch_bounds__(256) void wmma_gemm_lds_kernel(
        const float* __restrict__ A, const float* __restrict__ W,
        const float* __restrict__ bias, float* __restrict__ C, int M) {
    constexpr int NT = Nc / 16;
    __shared__ float sW[K * Nc];
    stage_to_lds<K * Nc>(sW, W);

    const int lane = threadIdx.x & 31;
    const int wave = threadIdx.x >> 5;
    const int half = lane >> 4;          // 0 or 1
    const int lm   = lane & 15;
    const int row0 = blockIdx.x * 128 + wave * 16;
    if (row0 >= M) return;               // wave-uniform: EXEC stays all-1s for WMMA

    const float* __restrict__ arow = A + (long long)(row0 + lm) * K;

    v8f acc[NT];
    const v8f vzero = {0.f, 0.f, 0.f, 0.f, 0.f, 0.f, 0.f, 0.f};
#pragma unroll
    for (int t = 0; t < NT; ++t) acc[t] = vzero;

    for (int k = 0; k < K; k += 4) {
        __builtin_prefetch(arow + k + 32, 0, 1);
        v2f a;                           // A 16x4 frag: VGPR v holds K = k + v + 2*half
        a.x = arow[k + 2 * half];
        a.y = arow[k + 2 * half + 1];
#pragma unroll
        for (int t = 0; t < NT; ++t) {
            v2f b;                       // B 4x16 frag from LDS
            b.x = sW[(k + 2 * half)     * Nc + t * 16 + lm];
            b.y = sW[(k + 2 * half + 1) * Nc + t * 16 + lm];
            acc[t] = __builtin_amdgcn_wmma_f32_16x16x4_f32(
                         false, a, false, b, (short)0, acc[t], false, false);
        }
    }
#pragma unroll
    for (int t = 0; t < NT; ++t) {
        const int col = t * 16 + lm;
        const float bs = bias[col];
#pragma unroll
        for (int v = 0; v < 8; ++v)      // C/D layout: M = v + 8*half
            C[(long long)(row0 + v + 8 * half) * Nc + col] = acc[t][v] + bs;
    }
}

// ---------------------------------------------------------------------------
// Per-edge GATv2 logits. One wave per head (wave32-native); each lane owns
// channels {lane, lane+32} of its head. Edge-attr broadcast via lane shuffles,
// logit reduced with shfl_xor — no __syncthreads in the edge loop.
//   m = leaky_relu(xl[src] + xr[dst] + edge_attr@We, 0.2)
//   logits[e,h] = dot(m[h], att[h]);  segment-max into mx[dst,h]
// ---------------------------------------------------------------------------
template <int H, int EPB>
__global__ void edge_logits_kernel(const float* __restrict__ XL,
                                   const float* __restrict__ XR,
                                   const float* __restrict__ EA,   // [E, EDIM]
                                   const float* __restrict__ We,   // [EDIM, H*64]
                                   const float* __restrict__ att,  // [H, 64]
                                   const int*   __restrict__ src,
                                   const int*   __restrict__ dst,
                                   float* __restrict__ logits,     // [E, H]
                                   float* __restrict__ mx,         // [N, H]
                                   int E) {
    constexpr int C  = 64;
    constexpr int NC = H * C;
    __shared__ float sWe[EDIM * NC];
    stage_to_lds<EDIM * NC>(sWe, We);

    const int lane = threadIdx.x & 31;
    const int h    = threadIdx.x >> 5;   // wave index == head index
    const int c0   = h * C + lane;
    const int c1   = c0 + 32;
    const float at0 = att[h * C + lane];
    const float at1 = att[h * C + lane + 32];

    int e0 = blockIdx.x * EPB;
    int e1 = e0 + EPB; if (e1 > E) e1 = E;
    for (int e = e0; e < e1; ++e) {
        const int s = src[e], d = dst[e];
        float eav = EA[(long long)e * EDIM + (lane & 15)];
        float ev0 = 0.f, ev1 = 0.f;
#pragma unroll
        for (int j = 0; j < EDIM; ++j) {
            float aj = __shfl(eav, j, 16);
            ev0 = fmaf(aj, sWe[j * NC + c0], ev0);
            ev1 = fmaf(aj, sWe[j * NC + c1], ev1);
        }
        float m0 = XL[(long long)s * NC + c0] + XR[(long long)d * NC + c0] + ev0;
        float m1 = XL[(long long)s * NC + c1] + XR[(long long)d * NC + c1] + ev1;
        m0 = (m0 > 0.f) ? m0 : 0.2f * m0;   // leaky_relu(0.2)
        m1 = (m1 > 0.f) ? m1 : 0.2f * m1;
        float p = fmaf(m0, at0, m1 * at1);
#pragma unroll
        for (int off = 16; off > 0; off >>= 1)
            p += __shfl_xor(p, off);
        if (lane == 0) {
            logits[(long long)e * H + h] = p;
            atomicMaxF(&mx[(long long)d * H + h], p);
        }
    }
}

// a[e,h] = exp(logits - mx[dst]); denom[dst,h] += a
__global__ void edge_softmax_kernel(const float* __restrict__ logits,
                                    const int*   __restrict__ dst,
                                    const float* __restrict__ mx,
                                    float* __restrict__ a,
                                    float* __restrict__ denom,
                                    int E, int H) {
    int i = blockIdx.x * blockDim.x + threadIdx.x;
    if (i >= E * H) return;
    int e = i / H, h = i % H;
    int d = dst[e];
    float v = __expf(logits[i] - mx[d * H + h]);
    a[i] = v;
    atomicAdd(&denom[d * H + h], v);
}

// out[dst] += xl[src] * alpha   (alpha = a / (denom[dst] + 1e-16))
__global__ void edge_aggregate_kernel(const float* __restrict__ XL,
                                      const float* __restrict__ a,
                                      const float* __restrict__ denom,
                                      const int* __restrict__ src,
                                      const int* __restrict__ dst,
                                      float* __restrict__ out,
                                      int E, int H, int C) {
    const int NC = H * C;
    long long total  = (long long)E * NC;
    long long i      = (long long)blockIdx.x * blockDim.x + threadIdx.x;
    long long stride = (long long)gridDim.x * blockDim.x;
    for (; i < total; i += stride) {
        int e = (int)(i / NC);
        int c = (int)(i % NC);
        int h = c / C;
        int d = dst[e];
        float alpha = a[(long long)e * H + h] / (denom[(long long)d * H + h] + 1e-16f);
        atomicAdd(&out[(long long)d * NC + c],
                  XL[(long long)src[e] * NC + c] * alpha);
    }
}

// ---------------------------------------------------------------------------
// BatchNorm stats (per-feature sum / sumsq over nodes), blockDim == F
// ---------------------------------------------------------------------------
__global__ void bn_stats_kernel(const float* __restrict__ X,
                                float* __restrict__ sum, float* __restrict__ sq,
                                int n, int F, int rpb) {
    const int f  = threadIdx.x;
    const int r0 = blockIdx.x * rpb;
    int r1 = r0 + rpb; if (r1 > n) r1 = n;
    float s = 0.f, ss = 0.f;
    for (int r = r0; r < r1; ++r) {
        float v = X[(long long)r * F + f];
        s += v; ss = fmaf(v, v, ss);
    }
    atomicAdd(&sum[f], s);
    atomicAdd(&sq[f],  ss);
}

// out = elu( (x-mean)*rsqrt(var+1e-5)*g + b + skip )
// (GATv2 output bias is a per-feature constant: it cancels inside BN's
//  (x - mean), so it is legitimately omitted before BN.)
__global__ void bn_elu_skip_kernel(const float* __restrict__ X,
                                   const float* __restrict__ sum,
                                   const float* __restrict__ sq,
                                   const float* __restrict__ g,
                                   const float* __restrict__ b,
                                   const float* __restrict__ skip,
                                   float* __restrict__ out,
                                   int n, int F) {
    long long i = (long long)blockIdx.x * blockDim.x + threadIdx.x;
    if (i >= (long long)n * F) return;
    int f = (int)(i % F);
    float inv_n = 1.f / (float)n;
    float mean  = sum[f] * inv_n;
    float var   = sq[f] * inv_n - mean * mean;
    float y = (X[i] - mean) * rsqrtf(var + 1e-5f) * g[f] + b[f] + skip[i];
    out[i] = (y > 0.f) ? y : (__expf(y) - 1.f);
}

// ---------------------------------------------------------------------------
// Pooling + classifier
// ---------------------------------------------------------------------------
__global__ void pool_count_kernel(const int* __restrict__ batch,
                                  float* __restrict__ counts, int n) {
    int i = blockIdx.x * blockDim.x + threadIdx.x;
    if (i < n) atomicAdd(&counts[batch[i]], 1.f);
}

__global__ void pool_kernel(const float* __restrict__ H2,
                            const int* __restrict__ batch,
                            float* __restrict__ psum, float* __restrict__ pmax,
                            int n, int F) {
    long long i = (long long)blockIdx.x * blockDim.x + threadIdx.x;
    if (i >= (long long)n * F) return;
    int node = (int)(i / F), f = (int)(i % F);
    int g = batch[node];
    float v = H2[i];
    atomicAdd(&psum[g * F + f], v);
    atomicMaxF(&pmax[g * F + f], v);
}

__global__ void classifier_kernel(const float* __restrict__ psum,
                                  const float* __restrict__ pmax,
                                  const float* __restrict__ counts,
                                  const float* __restrict__ Wc,  // [2*HID, NCLS]
                                  const float* __restrict__ bc,
                                  float* __restrict__ out) {
    int i = blockIdx.x * blockDim.x + threadIdx.x;
    if (i >= NGRAPH * NCLS) return;
    int g = i / NCLS, cls = i % NCLS;
    float cnt = fmaxf(counts[g], 1.f);
    float acc = bc[cls];
    for (int j = 0; j < HID; ++j)
        acc = fmaf(psum[g * HID + j] / cnt, Wc[j * NCLS + cls], acc);
    for (int j = 0; j < HID; ++j)
        acc = fmaf(pmax[g * HID + j], Wc[(HID + j) * NCLS + cls], acc);
    out[i] = acc;
}

// ---------------------------------------------------------------------------
// launch
// ---------------------------------------------------------------------------
extern "C" void kernel_launch(void* const* d_in, const int* in_sizes, int n_in,
                              void* d_out, int out_size, void* d_ws, size_t ws_size,
                              hipStream_t stream) {
    const float* x        = (const float*)d_in[0];
    const int*   ei       = (const int*)  d_in[1];
    const float* ea       = (const float*)d_in[2];
    const int*   batch    = (const int*)  d_in[3];
    const float* skip1_w  = (const float*)d_in[4];
    const float* skip1_b  = (const float*)d_in[5];
    const float* c1_wl    = (const float*)d_in[6];
    const float* c1_bl    = (const float*)d_in[7];
    const float* c1_wr    = (const float*)d_in[8];
    const float* c1_br    = (const float*)d_in[9];
    const float* c1_we    = (const float*)d_in[10];
    const float* c1_att   = (const float*)d_in[11];
    // d_in[12] = c1_bias (cancels inside BN; see bn_elu_skip_kernel comment)
    const float* bn1_g    = (const float*)d_in[13];
    const float* bn1_b    = (const float*)d_in[14];
    const float* skip2_w  = (const float*)d_in[15];
    const float* skip2_b  = (const float*)d_in[16];
    const float* c2_wl    = (const float*)d_in[17];
    const float* c2_bl    = (const float*)d_in[18];
    const float* c2_wr    = (const float*)d_in[19];
    const float* c2_br    = (const float*)d_in[20];
    const float* c2_we    = (const float*)d_in[21];
    const float* c2_att   = (const float*)d_in[22];
    // d_in[23] = c2_bias (cancels inside BN)
    const float* bn2_g    = (const float*)d_in[24];
    const float* bn2_b    = (const float*)d_in[25];
    const float* cls_w    = (const float*)d_in[26];
    const float* cls_b    = (const float*)d_in[27];

    const int* src = ei;
    const int* dst = ei + N_EDGES;

    float* ws = (float*)d_ws;
    size_t off = 0;
    auto alloc = [&](size_t nf) { float* p = ws + off; off += nf; return p; };

    float* XP1   = alloc((size_t)N_NODES * HC1);
    float* XL1   = alloc((size_t)N_NODES * HC1);
    float* XR1   = alloc((size_t)N_NODES * HC1);
    float* OUT1  = alloc((size_t)N_NODES * HC1);
    float* H1    = alloc((size_t)N_NODES * HC1);
    float* LG1   = alloc((size_t)N_EDGES * 2);
    float* A1    = alloc((size_t)N_EDGES * 2);
    float* MX1   = alloc((size_t)N_NODES * 2);
    float* DEN1  = alloc((size_t)N_NODES * 2);
    float* BS1   = alloc(HC1);
    float* BQ1   = alloc(HC1);
    float* XP2   = alloc((size_t)N_NODES * HID);
    float* XL2   = alloc((size_t)N_NODES * HID);
    float* XR2   = alloc((size_t)N_NODES * HID);
    float* OUT2  = alloc((size_t)N_NODES * HID);
    float* H2    = alloc((size_t)N_NODES * HID);
    float* LG2   = alloc((size_t)N_EDGES);
    float* A2    = alloc((size_t)N_EDGES);
    float* MX2   = alloc((size_t)N_NODES);
    float* DEN2  = alloc((size_t)N_NODES);
    float* BS2   = alloc(HID);
    float* BQ2   = alloc(HID);
    float* CNT   = alloc(64);
    float* PSUM  = alloc((size_t)NGRAPH * HID);
    float* PMAX  = alloc((size_t)NGRAPH * HID);

    const float NEG_INF = -INFINITY;
    auto fill = [&](float* p, float v, long long n) {
        int blocks = (int)((n + 255) / 256); if (blocks > 16384) blocks = 16384;
        fill_kernel<<<blocks, 256, 0, stream>>>(p, v, n);
    };

    const int gemm_blocks = (N_NODES + 127) / 128;   // 391 (last block partial-by-wave)

    // ---- Layer 1 node GEMMs (WMMA fp32, W staged via TDM/async LDS) ----
    wmma_gemm_lds_kernel<F_IN, HC1><<<gemm_blocks, 256, 0, stream>>>(x, c1_wl,   c1_bl,   XL1, N_NODES);
    wmma_gemm_lds_kernel<F_IN, HC1><<<gemm_blocks, 256, 0, stream>>>(x, c1_wr,   c1_br,   XR1, N_NODES);
    wmma_gemm_lds_kernel<F_IN, HC1><<<gemm_blocks, 256, 0, stream>>>(x, skip1_w, skip1_b, XP1, N_NODES);

    // ---- Layer 1 attention ----
    fill(MX1, NEG_INF, (long long)N_NODES * 2);
    fill(DEN1, 0.f,    (long long)N_NODES * 2);
    fill(OUT1, 0.f,    (long long)N_NODES * HC1);
    edge_logits_kernel<2, 16><<<(N_EDGES + 15) / 16, 64, 0, stream>>>(
        XL1, XR1, ea, c1_we, c1_att, src, dst, LG1, MX1, N_EDGES);
    edge_softmax_kernel<<<(N_EDGES * 2 + 255) / 256, 256, 0, stream>>>(
        LG1, dst, MX1, A1, DEN1, N_EDGES, 2);
    edge_aggregate_kernel<<<200000, 256, 0, stream>>>(
        XL1, A1, DEN1, src, dst, OUT1, N_EDGES, 2, 64);

    // ---- BN1 + ELU + skip ----
    fill(BS1, 0.f, HC1);
    fill(BQ1, 0.f, HC1);
    bn_stats_kernel<<<(N_NODES + 127) / 128, HC1, 0, stream>>>(OUT1, BS1, BQ1, N_NODES, HC1, 128);
    bn_elu_skip_kernel<<<(int)(((long long)N_NODES * HC1 + 255) / 256), 256, 0, stream>>>(
        OUT1, BS1, BQ1, bn1_g, bn1_b, XP1, H1, N_NODES, HC1);

    // ---- Layer 2 node GEMMs ----
    wmma_gemm_lds_kernel<HC1, HID><<<gemm_blocks, 256, 0, stream>>>(H1, c2_wl,   c2_bl,   XL2, N_NODES);
    wmma_gemm_lds_kernel<HC1, HID><<<gemm_blocks, 256, 0, stream>>>(H1, c2_wr,   c2_br,   XR2, N_NODES);
    wmma_gemm_lds_kernel<HC1, HID><<<gemm_blocks, 256, 0, stream>>>(H1, skip2_w, skip2_b, XP2, N_NODES);

    // ---- Layer 2 attention ----
    fill(MX2, NEG_INF, N_NODES);
    fill(DEN2, 0.f,    N_NODES);
    fill(OUT2, 0.f,    (long long)N_NODES * HID);
    edge_logits_kernel<1, 16><<<(N_EDGES + 15) / 16, 32, 0, stream>>>(
        XL2, XR2, ea, c2_we, c2_att, src, dst, LG2, MX2, N_EDGES);
    edge_softmax_kernel<<<(N_EDGES + 255) / 256, 256, 0, stream>>>(
        LG2, dst, MX2, A2, DEN2, N_EDGES, 1);
    edge_aggregate_kernel<<<100000, 256, 0, stream>>>(
        XL2, A2, DEN2, src, dst, OUT2, N_EDGES, 1, 64);

    // ---- BN2 + ELU + skip ----
    fill(BS2, 0.f, HID);
    fill(BQ2, 0.f, HID);
    bn_stats_kernel<<<(N_NODES + 127) / 128, HID, 0, stream>>>(OUT2, BS2, BQ2, N_NODES, HID, 128);
    bn_elu_skip_kernel<<<(int)(((long long)N_NODES * HID + 255) / 256), 256, 0, stream>>>(
        OUT2, BS2, BQ2, bn2_g, bn2_b, XP2, H2, N_NODES, HID);

    // ---- Pooling + classifier ----
    fill(CNT, 0.f, 64);
    fill(PSUM, 0.f, (long long)NGRAPH * HID);
    fill(PMAX, NEG_INF, (long long)NGRAPH * HID);
    pool_count_kernel<<<(N_NODES + 255) / 256, 256, 0, stream>>>(batch, CNT, N_NODES);
    pool_kernel<<<(int)(((long long)N_NODES * HID + 255) / 256), 256, 0, stream>>>(
        H2, batch, PSUM, PMAX, N_NODES, HID);
    classifier_kernel<<<1, 128, 0, stream>>>(PSUM, PMAX, CNT, cls_w, cls_b, (float*)d_out);
}